// AttentionEncoderDecoderModel_76630806496015
// MI455X (gfx1250) — compile-verified
//
#include <hip/hip_runtime.h>
#include <hip/hip_bf16.h>
#include <math.h>

// ---------------- CDNA5 WMMA fp32 path ----------------
typedef __attribute__((ext_vector_type(2))) float v2f;
typedef __attribute__((ext_vector_type(4))) float v4f;
typedef __attribute__((ext_vector_type(8))) float v8f;

__device__ __forceinline__ v8f v8zero() {
    v8f v;
#pragma unroll
    for (int i = 0; i < 8; ++i) v[i] = 0.f;
    return v;
}

// D = A(16x4) * B(4x16) + C, exact fp32 via V_WMMA_F32_16X16X4_F32
__device__ __forceinline__ v8f wmma4(v2f a, v2f b, v8f c) {
    return __builtin_amdgcn_wmma_f32_16x16x4_f32(
        false, a, false, b, (short)0, c, false, false);
}

__device__ __forceinline__ float sigmf(float x) { return 1.f / (1.f + expf(-x)); }

// Model dims
#define BB 32
#define TT 128
#define TY 64
#define EE 512
#define HH2 256   /* per-direction encoder hidden */
#define HD 512    /* decoder hidden */
#define VV 32000

// ---------------- K1: pad length + reversed token indices ----------------
__global__ void prep_kernel(const int* __restrict__ x, const float* __restrict__ xm,
                            int* __restrict__ xback) {
    __shared__ float red[TT];
    const int b = blockIdx.x, t = threadIdx.x;
    red[t] = xm[b * TT + t];
    __syncthreads();
    for (int s = TT / 2; s > 0; s >>= 1) {
        if (t < s) red[t] += red[t + s];
        __syncthreads();
    }
    const int pad = TT - (int)(red[0] + 0.5f);
    const int j = TT - 1 - pad - t;  // inverse of rev scatter (exact for pad==0)
    xback[b * TT + t] = (j >= 0) ? x[b * TT + j] : 0;
}

// ---------------- K2: generic WMMA GEMM  G[m,n] = sum_k A[m,k]*W[n,k] + b1[n]+b2[n]
// A row = (tok ? embed[tok[m]] : A + m*K). Wave strip = 16(M) x 64(N).
// K-chunked: 20 loads batched ahead of 16 WMMAs per chunk to keep loadcnt deep.
__global__ void gemm16_kernel(const float* __restrict__ A,
                              const int* __restrict__ tok,
                              const float* __restrict__ embedTab,
                              const float* __restrict__ W,
                              const float* __restrict__ bias1,
                              const float* __restrict__ bias2,
                              float* __restrict__ G,
                              int M, int N, int K) {
    const int lane = threadIdx.x & 31;
    const int wave = (int)((blockIdx.x * blockDim.x + threadIdx.x) >> 5);
    const int stripsN = N >> 6;
    const int totalStrips = (M >> 4) * stripsN;
    if (wave >= totalStrips) return;  // wave-uniform: EXEC stays all-ones for WMMA
    const int m0 = (wave / stripsN) << 4;
    const int n0 = (wave % stripsN) << 6;
    const int col = lane & 15;
    const int hi = lane >> 4;

    const float* Arow = tok ? (embedTab + (size_t)tok[m0 + col] * K)
                            : (A + (size_t)(m0 + col) * K);
    const float* Wp[4];
#pragma unroll
    for (int nt = 0; nt < 4; ++nt) Wp[nt] = W + (size_t)(n0 + (nt << 4) + col) * K;

    v8f c[4];
#pragma unroll
    for (int nt = 0; nt < 4; ++nt) c[nt] = v8zero();

    for (int k0 = 0; k0 < K; k0 += 16) {
        v2f a[4], bv[4][4];
#pragma unroll
        for (int u = 0; u < 4; ++u) {
            const int ka = k0 + (u << 2) + (hi << 1);
            a[u].x = Arow[ka];
            a[u].y = Arow[ka + 1];
#pragma unroll
            for (int nt = 0; nt < 4; ++nt) {
                bv[u][nt].x = Wp[nt][ka];
                bv[u][nt].y = Wp[nt][ka + 1];
            }
        }
#pragma unroll
        for (int u = 0; u < 4; ++u)
#pragma unroll
            for (int nt = 0; nt < 4; ++nt) c[nt] = wmma4(a[u], bv[u][nt], c[nt]);
    }
#pragma unroll
    for (int nt = 0; nt < 4; ++nt) {
        const int n = n0 + (nt << 4) + col;
        float bs = bias1 ? bias1[n] : 0.f;
        if (bias2) bs += bias2[n];
#pragma unroll
        for (int r = 0; r < 8; ++r) {
            const int mm = m0 + r + (hi << 3);
            G[(size_t)mm * N + n] = c[nt][r] + bs;
        }
    }
}

// ---------------- K6: transposed vocab projection ----------------
// logits[n, v] = sum_k outh[n,k] * W_out[v,k] + b_out[v]
// A = W_out rows (vocab = M, streamed from HBM exactly once);
// B = outh rows (batch = N, L2-resident, re-read). D stored transposed ->
// contiguous along vocab => two aligned b128 stores per lane per tile.
__global__ void projT_kernel(const float* __restrict__ Wout,   // (V, K)
                             const float* __restrict__ Bmat,   // (N=B*TY, K)
                             const float* __restrict__ bias,   // (V)
                             float* __restrict__ logits) {     // (N, V)
    const int lane = threadIdx.x & 31;
    const int wave = (int)((blockIdx.x * blockDim.x + threadIdx.x) >> 5);
    const int stripsN = (BB * TY) >> 6;           // 32
    const int totalStrips = (VV >> 4) * stripsN;  // 64000
    if (wave >= totalStrips) return;
    const int m0 = (wave / stripsN) << 4;  // vocab tile base
    const int n0 = (wave % stripsN) << 6;  // batch tile base
    const int col = lane & 15;
    const int hi = lane >> 4;
    const int K = 2 * HD;

    const float* Arow = Wout + (size_t)(m0 + col) * K;
    const float* Bp[4];
#pragma unroll
    for (int nt = 0; nt < 4; ++nt) Bp[nt] = Bmat + (size_t)(n0 + (nt << 4) + col) * K;

    v8f c[4];
#pragma unroll
    for (int nt = 0; nt < 4; ++nt) c[nt] = v8zero();

    for (int k0 = 0; k0 < K; k0 += 16) {
        v2f a[4], bv[4][4];
#pragma unroll
        for (int u = 0; u < 4; ++u) {
            const int ka = k0 + (u << 2) + (hi << 1);
            a[u].x = Arow[ka];
            a[u].y = Arow[ka + 1];
#pragma unroll
            for (int nt = 0; nt < 4; ++nt) {
                bv[u][nt].x = Bp[nt][ka];
                bv[u][nt].y = Bp[nt][ka + 1];
            }
        }
        // stream-ahead hint for the once-read W_out rows (global_prefetch_b8)
        __builtin_prefetch(Arow + k0 + 32, 0, 0);
#pragma unroll
        for (int u = 0; u < 4; ++u)
#pragma unroll
            for (int nt = 0; nt < 4; ++nt) c[nt] = wmma4(a[u], bv[u][nt], c[nt]);
    }

    const v4f blo = *(const v4f*)(bias + m0 + (hi << 3));
    const v4f bhi = *(const v4f*)(bias + m0 + (hi << 3) + 4);
#pragma unroll
    for (int nt = 0; nt < 4; ++nt) {
        const int n = n0 + (nt << 4) + col;
        float* dp = logits + (size_t)n * VV + m0 + (hi << 3);
        v4f lo, hi4;
#pragma unroll
        for (int i = 0; i < 4; ++i) {
            lo[i] = c[nt][i] + blo[i];
            hi4[i] = c[nt][4 + i] + bhi[i];
        }
        *(v4f*)dp = lo;
        *(v4f*)(dp + 4) = hi4;
    }
}

// ---------------- K3: persistent bidirectional encoder (grid=2, block=512=16 waves)
__global__ void encoder_kernel(const float* __restrict__ xg_f, const float* __restrict__ xg_b,
                               const float* __restrict__ whh_f, const float* __restrict__ whh_b,
                               float* __restrict__ hiddens, float* __restrict__ cells) {
    const int dir = blockIdx.x;
    const float* xg = dir ? xg_b : xg_f;     // (B*T, 4*HH2), x@Wih^T + b_ih + b_hh
    const float* whh = dir ? whh_b : whh_f;  // (4*HH2, HH2)

    __shared__ float h_sh[BB][HH2 + 1];  // stride 257 words -> conflict-free

    const int lane = threadIdx.x & 31;
    const int wave = threadIdx.x >> 5;  // 16 waves -> hidden chunk of 16
    const int j0 = wave << 4;
    const int col = lane & 15;
    const int hi = lane >> 4;

    for (int i = threadIdx.x; i < BB * (HH2 + 1); i += blockDim.x)
        ((float*)h_sh)[i] = 0.f;
    v8f cc[2];
    cc[0] = v8zero();
    cc[1] = v8zero();
    __syncthreads();

    for (int t = 0; t < TT; ++t) {
        v8f acc[4][2];
#pragma unroll
        for (int g = 0; g < 4; ++g) { acc[g][0] = v8zero(); acc[g][1] = v8zero(); }

        for (int k0 = 0; k0 < HH2; k0 += 16) {
            v2f a0[4], a1[4], bv[4][4];
#pragma unroll
            for (int u = 0; u < 4; ++u) {
                const int ka = k0 + (u << 2) + (hi << 1);
                a0[u].x = h_sh[col][ka];      a0[u].y = h_sh[col][ka + 1];
                a1[u].x = h_sh[16 + col][ka]; a1[u].y = h_sh[16 + col][ka + 1];
#pragma unroll
                for (int g = 0; g < 4; ++g) {
                    const float* wp = whh + (size_t)((g << 8) + j0 + col) * HH2 + ka;
                    bv[u][g].x = wp[0];
                    bv[u][g].y = wp[1];
                }
            }
#pragma unroll
            for (int u = 0; u < 4; ++u)
#pragma unroll
                for (int g = 0; g < 4; ++g) {
                    acc[g][0] = wmma4(a0[u], bv[u][g], acc[g][0]);
                    acc[g][1] = wmma4(a1[u], bv[u][g], acc[g][1]);
                }
        }
        __syncthreads();  // all waves done reading old h

        const int j = j0 + col;
#pragma unroll
        for (int mt = 0; mt < 2; ++mt) {
#pragma unroll
            for (int r = 0; r < 8; ++r) {
                const int bb = (mt << 4) + r + (hi << 3);
                const float* xr = xg + ((size_t)bb * TT + t) * (4 * HH2);
                const float gi = acc[0][mt][r] + xr[j];
                const float gf = acc[1][mt][r] + xr[HH2 + j];
                const float gg = acc[2][mt][r] + xr[2 * HH2 + j];
                const float go = acc[3][mt][r] + xr[3 * HH2 + j];
                const float cv = sigmf(gf) * cc[mt][r] + sigmf(gi) * tanhf(gg);
                cc[mt][r] = cv;
                const float hv = sigmf(go) * tanhf(cv);
                h_sh[bb][j] = hv;
                const size_t o = ((size_t)bb * TT + t) * (2 * HH2) + dir * HH2 + j;
                hiddens[o] = hv;
                cells[o] = cv;
            }
        }
        __syncthreads();
    }
}

// ---------------- K4: mean over T ----------------
__global__ void mean_kernel(const float* __restrict__ hiddens, const float* __restrict__ cells,
                            float* __restrict__ h_mean, float* __restrict__ c_mean) {
    const int i = blockIdx.x * blockDim.x + threadIdx.x;
    if (i >= BB * HD) return;
    const int bb = i >> 9, hh = i & 511;
    float sh = 0.f, sc = 0.f;
    for (int t = 0; t < TT; ++t) {
        const size_t o = ((size_t)bb * TT + t) * HD + hh;
        sh += hiddens[o];
        sc += cells[o];
    }
    h_mean[i] = sh * (1.f / (float)TT);
    c_mean[i] = sc * (1.f / (float)TT);
}

// ---------------- K5: persistent decoder + attention (grid=1, block=1024=32 waves)
__global__ void decoder_kernel(const float* __restrict__ yg,   // (B*TY, 4*HD) precomputed
                               const float* __restrict__ whh,  // (4*HD, HD)
                               const float* __restrict__ h_mean, const float* __restrict__ c_mean,
                               const float* __restrict__ hiddens,  // (B,T,HD)
                               float* __restrict__ outh) {          // (B*TY, 2*HD)
    __shared__ float h_sh[BB][HD + 4];  // 4-bank lane stride, conflict-free

    const int lane = threadIdx.x & 31;
    const int wave = threadIdx.x >> 5;  // hidden chunk of 16; also batch id for attention
    const int j0 = wave << 4;
    const int col = lane & 15;
    const int hi = lane >> 4;

    for (int i = threadIdx.x; i < BB * HD; i += blockDim.x)
        h_sh[i >> 9][i & 511] = h_mean[i];

    v8f cc[2];
#pragma unroll
    for (int mt = 0; mt < 2; ++mt) {
#pragma unroll
        for (int r = 0; r < 8; ++r) {
            const int bb = (mt << 4) + r + (hi << 3);
            cc[mt][r] = c_mean[(size_t)bb * HD + j0 + col];
        }
    }
    __syncthreads();

    for (int ty = 0; ty < TY; ++ty) {
        // --- recurrent gates: R = h @ whh^T on the wave's 16 hidden columns ---
        v8f acc[4][2];
#pragma unroll
        for (int g = 0; g < 4; ++g) { acc[g][0] = v8zero(); acc[g][1] = v8zero(); }

        for (int k0 = 0; k0 < HD; k0 += 16) {
            v2f a0[4], a1[4], bv[4][4];
#pragma unroll
            for (int u = 0; u < 4; ++u) {
                const int ka = k0 + (u << 2) + (hi << 1);
                a0[u].x = h_sh[col][ka];      a0[u].y = h_sh[col][ka + 1];
                a1[u].x = h_sh[16 + col][ka]; a1[u].y = h_sh[16 + col][ka + 1];
#pragma unroll
                for (int g = 0; g < 4; ++g) {
                    const float* wp = whh + (size_t)((g << 9) + j0 + col) * HD + ka;
                    bv[u][g].x = wp[0];
                    bv[u][g].y = wp[1];
                }
            }
#pragma unroll
            for (int u = 0; u < 4; ++u)
#pragma unroll
                for (int g = 0; g < 4; ++g) {
                    acc[g][0] = wmma4(a0[u], bv[u][g], acc[g][0]);
                    acc[g][1] = wmma4(a1[u], bv[u][g], acc[g][1]);
                }
        }
        __syncthreads();  // done reading old h

        const int j = j0 + col;
#pragma unroll
        for (int mt = 0; mt < 2; ++mt) {
#pragma unroll
            for (int r = 0; r < 8; ++r) {
                const int bb = (mt << 4) + r + (hi << 3);
                const float* yr = yg + ((size_t)bb * TY + ty) * (4 * HD);
                const float gi = acc[0][mt][r] + yr[j];
                const float gf = acc[1][mt][r] + yr[HD + j];
                const float gg = acc[2][mt][r] + yr[2 * HD + j];
                const float go = acc[3][mt][r] + yr[3 * HD + j];
                const float cv = sigmf(gf) * cc[mt][r] + sigmf(gi) * tanhf(gg);
                cc[mt][r] = cv;
                const float hv = sigmf(go) * tanhf(cv);
                h_sh[bb][j] = hv;
                outh[((size_t)bb * TY + ty) * (2 * HD) + j] = hv;  // first half = h
            }
        }
        __syncthreads();  // new h visible

        // --- dot attention: wave w handles batch b = w (wave32 shuffle softmax) ---
        const float* hb = hiddens + (size_t)wave * TT * HD;
        const float* hq = &h_sh[wave][0];
        float s[4];
#pragma unroll
        for (int q = 0; q < 4; ++q) {
            const float* hr = hb + (size_t)(lane + (q << 5)) * HD;
            float d = 0.f;
            for (int k = 0; k < HD; ++k) d += hr[k] * hq[k];
            s[q] = d;
        }
        float mx = fmaxf(fmaxf(s[0], s[1]), fmaxf(s[2], s[3]));
        for (int off = 16; off > 0; off >>= 1) mx = fmaxf(mx, __shfl_xor(mx, off, 32));
        float e[4];
        float se = 0.f;
#pragma unroll
        for (int q = 0; q < 4; ++q) { e[q] = expf(s[q] - mx); se += e[q]; }
        for (int off = 16; off > 0; off >>= 1) se += __shfl_xor(se, off, 32);
        const float inv = 1.f / se;

        float ctx[16];
#pragma unroll
        for (int ci = 0; ci < 16; ++ci) ctx[ci] = 0.f;
#pragma unroll
        for (int q = 0; q < 4; ++q) {
            for (int t2 = 0; t2 < 32; ++t2) {
                const float at = __shfl(e[q], t2, 32) * inv;
                const float* hr = hb + (size_t)((q << 5) + t2) * HD + (lane << 4);
#pragma unroll
                for (int ci = 0; ci < 16; ++ci) ctx[ci] += at * hr[ci];
            }
        }
        float* op = outh + ((size_t)wave * TY + ty) * (2 * HD) + HD + (lane << 4);
#pragma unroll
        for (int ci = 0; ci < 16; ++ci) op[ci] = ctx[ci];
        // next iteration's GEMM only reads h_sh; its post-GEMM barrier orders rewrites
    }
}

// ---------------- K7: in-place log_softmax over V per row ----------------
__global__ void logsoftmax_kernel(float* __restrict__ logits, int N) {
    __shared__ float red[256];
    float* p = logits + (size_t)blockIdx.x * N;
    const int tid = threadIdx.x;

    float m = -3.0e38f;
    for (int i = tid; i < N; i += 256) m = fmaxf(m, p[i]);
    red[tid] = m;
    __syncthreads();
    for (int s = 128; s > 0; s >>= 1) {
        if (tid < s) red[tid] = fmaxf(red[tid], red[tid + s]);
        __syncthreads();
    }
    m = red[0];
    __syncthreads();

    float sum = 0.f;
    for (int i = tid; i < N; i += 256) sum += expf(p[i] - m);
    red[tid] = sum;
    __syncthreads();
    for (int s = 128; s > 0; s >>= 1) {
        if (tid < s) red[tid] += red[tid + s];
        __syncthreads();
    }
    const float lse = m + logf(red[0]);
    __syncthreads();

    for (int i = tid; i < N; i += 256) p[i] = p[i] - lse;
}

// ---------------- host side ----------------
extern "C" void kernel_launch(void* const* d_in, const int* in_sizes, int n_in,
                              void* d_out, int out_size, void* d_ws, size_t ws_size,
                              hipStream_t stream) {
    const int* x        = (const int*)d_in[0];
    const float* x_mask = (const float*)d_in[1];
    const int* y        = (const int*)d_in[2];
    const float* embed  = (const float*)d_in[3];
    const float* w_ih_f = (const float*)d_in[4];
    const float* w_hh_f = (const float*)d_in[5];
    const float* b_ih_f = (const float*)d_in[6];
    const float* b_hh_f = (const float*)d_in[7];
    const float* w_ih_b = (const float*)d_in[8];
    const float* w_hh_b = (const float*)d_in[9];
    const float* b_ih_b = (const float*)d_in[10];
    const float* b_hh_b = (const float*)d_in[11];
    const float* w_ih_d = (const float*)d_in[12];
    const float* w_hh_d = (const float*)d_in[13];
    const float* b_ih_d = (const float*)d_in[14];
    const float* b_hh_d = (const float*)d_in[15];
    const float* W_out  = (const float*)d_in[16];
    const float* b_out  = (const float*)d_in[17];

    // workspace layout (floats)
    float* ws = (float*)d_ws;
    int* xback   = (int*)d_ws;                       // B*T = 4096 ints
    float* xg_f  = ws + 4096;                        // 4096*1024
    float* xg_b  = xg_f + (size_t)4096 * 1024;       // 4096*1024
    float* yg_d  = xg_b + (size_t)4096 * 1024;       // 2048*2048
    float* hidd  = yg_d + (size_t)2048 * 2048;       // 4096*512
    float* cel   = hidd + (size_t)4096 * 512;        // 4096*512
    float* h_mean = cel + (size_t)4096 * 512;        // 16384
    float* c_mean = h_mean + 16384;                  // 16384

    float* logits = (float*)d_out;                   // (B*TY, V)
    float* outh   = logits + (size_t)BB * TY * VV;   // (B*TY, 2H)

    // 1) pad-aware reversal indices
    prep_kernel<<<BB, TT, 0, stream>>>(x, x_mask, xback);

    // 2) hoisted input-gate GEMMs (embedding gather fused into A loads)
    gemm16_kernel<<<512, 256, 0, stream>>>(nullptr, x, embed, w_ih_f, b_ih_f, b_hh_f,
                                           xg_f, BB * TT, 4 * HH2, EE);
    gemm16_kernel<<<512, 256, 0, stream>>>(nullptr, xback, embed, w_ih_b, b_ih_b, b_hh_b,
                                           xg_b, BB * TT, 4 * HH2, EE);
    gemm16_kernel<<<512, 256, 0, stream>>>(nullptr, y, embed, w_ih_d, b_ih_d, b_hh_d,
                                           yg_d, BB * TY, 4 * HD, EE);

    // 3) bidirectional LSTM encoder (persistent, WMMA recurrence)
    encoder_kernel<<<2, 512, 0, stream>>>(xg_f, xg_b, w_hh_f, w_hh_b, hidd, cel);

    // 4) means of hiddens/cells over T
    mean_kernel<<<(BB * HD + 255) / 256, 256, 0, stream>>>(hidd, cel, h_mean, c_mean);

    // 5) decoder LSTM + dot attention (persistent, WMMA recurrence)
    decoder_kernel<<<1, 1024, 0, stream>>>(yg_d, w_hh_d, h_mean, c_mean, hidd, outh);

    // 6) vocab projection, transposed mapping: W_out streamed once from HBM,
    //    outh L2-resident as B operand; contiguous b128 stores along vocab.
    projT_kernel<<<8000, 256, 0, stream>>>(W_out, outh, b_out, logits);

    // 7) in-place log_softmax over vocab
    logsoftmax_kernel<<<BB * TY, 256, 0, stream>>>(logits, VV);
}